// SAGE_50714973831907
// MI455X (gfx1250) — compile-verified
//
#include <hip/hip_runtime.h>

// ---------------------------------------------------------------------------
// GraphSAGE (3-layer, mean aggregation) for MI455X / gfx1250, wave32.
// fp32 end-to-end; GEMMs use V_WMMA_F32_16X16X4_F32 (exact fp32 matrix pipe).
// Workload is L2-atomic / memory bound, so fp32 WMMA costs nothing vs bf16.
// ---------------------------------------------------------------------------

#define N_NODES 100000
#define N_EDGES 600000
#define FEATS   128

typedef __attribute__((ext_vector_type(2))) float v2f;
typedef __attribute__((ext_vector_type(8))) float v8f;

// ---------------------------------------------------------------------------
__global__ void fill_zero_f32(float* __restrict__ p, int n) {
    int i = blockIdx.x * blockDim.x + threadIdx.x;
    int stride = gridDim.x * blockDim.x;
    for (; i < n; i += stride) p[i] = 0.0f;
}

__global__ void deg_count(const int* __restrict__ dst, float* __restrict__ deg,
                          int n_edges) {
    int e = blockIdx.x * blockDim.x + threadIdx.x;
    if (e < n_edges) atomicAdd(&deg[dst[e]], 1.0f);
}

__global__ void inv_deg_inplace(float* __restrict__ deg, int n) {
    int i = blockIdx.x * blockDim.x + threadIdx.x;
    if (i < n) deg[i] = 1.0f / fmaxf(deg[i], 1.0f);
}

// One wave per edge: lane l moves h[src, 4l..4l+3] into agg[dst, 4l..4l+3]
// via global_atomic_add_f32 (L2-resident; agg fits in the 192MB L2).
__global__ void sage_scatter(const float* __restrict__ h,
                             const int* __restrict__ src,
                             const int* __restrict__ dst,
                             float* __restrict__ agg, int n_edges) {
    int t = blockIdx.x * blockDim.x + threadIdx.x;
    int e = t >> 5;
    int lane = t & 31;
    if (e >= n_edges) return;
    int s = src[e];
    int d = dst[e];
    const float4 v = *((const float4*)(h + (size_t)s * FEATS) + lane);
    float* ap = agg + (size_t)d * FEATS + (size_t)lane * 4;
    atomicAdd(ap + 0, v.x);
    atomicAdd(ap + 1, v.y);
    atomicAdd(ap + 2, v.z);
    atomicAdd(ap + 3, v.w);
}

// ---------------------------------------------------------------------------
// Fused SAGE layer GEMM:  out = h @ Wself^T + (agg*inv_deg) @ Wneigh^T + b
// Block = (dout/16) waves, each wave owns one 16x16 output tile of a 16-row
// node group. K accumulated 4 at a time with V_WMMA_F32_16X16X4_F32.
//
// A-fragment layout (16x4 f32):  lane l holds {A[l&15, k0+2*(l>>4)], +1}
// B-fragment layout (4x16 f32):  lane l holds {B[2*(l>>4), l&15], B[.+1, .]}
//   with B = W^T, i.e. {W[ncol, k0+2*(l>>4)], W[ncol, k0+2*(l>>4)+1]}
// => both fragments are one 8-byte load per lane (256B/fragment, exact).
//
// __syncthreads() before the epilogue makes out==h (in-place) safe: all
// reads of this block's 16 h-rows complete before any wave writes them.
__global__ void __launch_bounds__(256)
sage_gemm(const float* h, const float* __restrict__ agg,
          const float* __restrict__ inv_deg,
          const float* __restrict__ Wself, const float* __restrict__ Wneigh,
          const float* __restrict__ bias, float* out,
          int dout, int relu) {
    const int lane  = threadIdx.x & 31;
    const int wave  = threadIdx.x >> 5;
    const int row0  = blockIdx.x << 4;
    const int m     = lane & 15;        // A row within tile
    const int khalf = lane >> 4;        // K sub-pair select (0 or 1)
    const int ncol  = (wave << 4) + (lane & 15);  // global output column

    const float* Ah = h    + (size_t)(row0 + m) * FEATS + 2 * khalf;
    const float* An = agg  + (size_t)(row0 + m) * FEATS + 2 * khalf;
    const float* Bs = Wself  + (size_t)ncol * FEATS + 2 * khalf;
    const float* Bn = Wneigh + (size_t)ncol * FEATS + 2 * khalf;
    const float  s  = inv_deg[row0 + m];

    v8f acc = {};

    // self contribution: K = 0..127
#pragma unroll 8
    for (int k = 0; k < FEATS; k += 4) {
        v2f a = *(const v2f*)(Ah + k);
        v2f b = *(const v2f*)(Bs + k);
        acc = __builtin_amdgcn_wmma_f32_16x16x4_f32(
            false, a, false, b, (short)0, acc, false, false);
    }
    // neighbor contribution (mean = agg * inv_deg): K = 0..127
#pragma unroll 8
    for (int k = 0; k < FEATS; k += 4) {
        v2f a = *(const v2f*)(An + k);
        a.x *= s;
        a.y *= s;
        v2f b = *(const v2f*)(Bn + k);
        acc = __builtin_amdgcn_wmma_f32_16x16x4_f32(
            false, a, false, b, (short)0, acc, false, false);
    }

    __syncthreads();  // enables in-place out==h across waves of this block

    // C/D layout: VGPR v, lane l -> row = v + 8*(l>>4), col = l&15
    const float bv = bias[ncol];
#pragma unroll
    for (int v = 0; v < 8; ++v) {
        float r = acc[v] + bv;
        if (relu) r = fmaxf(r, 0.0f);
        out[(size_t)(row0 + v + 8 * khalf) * dout + ncol] = r;
    }
}

// ---------------------------------------------------------------------------
extern "C" void kernel_launch(void* const* d_in, const int* in_sizes, int n_in,
                              void* d_out, int out_size, void* d_ws,
                              size_t ws_size, hipStream_t stream) {
    (void)in_sizes; (void)n_in; (void)out_size; (void)ws_size;

    const float* x    = (const float*)d_in[0];
    const int*   esrc = (const int*)d_in[1];
    const int*   edst = (const int*)d_in[2];
    const float* Ws1  = (const float*)d_in[3];
    const float* Wn1  = (const float*)d_in[4];
    const float* b1   = (const float*)d_in[5];
    const float* Ws2  = (const float*)d_in[6];
    const float* Wn2  = (const float*)d_in[7];
    const float* b2   = (const float*)d_in[8];
    const float* Ws3  = (const float*)d_in[9];
    const float* Wn3  = (const float*)d_in[10];
    const float* b3   = (const float*)d_in[11];
    float* out = (float*)d_out;

    // workspace layout (floats):
    //   deg/inv_deg : [0, N)           (padded to 131072 for alignment)
    //   agg         : N*128            (51.2 MB)
    //   h           : N*128            (51.2 MB, layer-2 runs in-place)
    float* ws  = (float*)d_ws;
    float* deg = ws;
    float* agg = ws + (size_t)131072;
    float* h   = agg + (size_t)N_NODES * FEATS;

    const int ZB = 2048, TB = 256;
    const int edge_blocks    = (N_EDGES + TB - 1) / TB;
    const int scatter_blocks = (N_EDGES * 32) / TB;   // 600000*32/256 = 75000
    const int gemm_blocks    = N_NODES / 16;          // 6250, exact

    // degree -> inv_deg (once, reused by all layers)
    fill_zero_f32<<<ZB, TB, 0, stream>>>(deg, N_NODES);
    deg_count<<<edge_blocks, TB, 0, stream>>>(edst, deg, N_EDGES);
    inv_deg_inplace<<<(N_NODES + TB - 1) / TB, TB, 0, stream>>>(deg, N_NODES);

    // ---- layer 1: x -> h (dout=128, relu) ----
    fill_zero_f32<<<ZB, TB, 0, stream>>>(agg, N_NODES * FEATS);
    sage_scatter<<<scatter_blocks, TB, 0, stream>>>(x, esrc, edst, agg, N_EDGES);
    sage_gemm<<<gemm_blocks, 256, 0, stream>>>(x, agg, deg, Ws1, Wn1, b1, h,
                                               128, 1);

    // ---- layer 2: h -> h (in-place, dout=128, relu) ----
    fill_zero_f32<<<ZB, TB, 0, stream>>>(agg, N_NODES * FEATS);
    sage_scatter<<<scatter_blocks, TB, 0, stream>>>(h, esrc, edst, agg, N_EDGES);
    sage_gemm<<<gemm_blocks, 256, 0, stream>>>(h, agg, deg, Ws2, Wn2, b2, h,
                                               128, 1);

    // ---- layer 3: h -> out (dout=64, no relu; 4 waves/block) ----
    fill_zero_f32<<<ZB, TB, 0, stream>>>(agg, N_NODES * FEATS);
    sage_scatter<<<scatter_blocks, TB, 0, stream>>>(h, esrc, edst, agg, N_EDGES);
    sage_gemm<<<gemm_blocks, 128, 0, stream>>>(h, agg, deg, Ws3, Wn3, b3, out,
                                               64, 0);
}